// MultBoxLoss_56994216018023
// MI455X (gfx1250) — compile-verified
//
#include <hip/hip_runtime.h>
#include <hip/hip_bf16.h>
#include <stdint.h>

// ---------------- problem constants ----------------
#define BB 64
#define PP 24564
#define CC 81
#define TT 50
#define THRESH 0.5f
#define VAR0 0.1f
#define VAR1 0.2f

#define NBLKX 24          // chunks per batch for match/ce kernels
#define CE_ROWS 128       // rows (priors) per CE tile
#define CE_TILES 192      // ceil(PP / CE_ROWS)
#define CE_TILES_PER_BLK 8// CE_TILES / NBLKX
#define CE_NLOAD 21       // ceil(CE_ROWS*81/4 / 128) async b128 issues per tile per thread
#define CE_CAP16 2688     // padded float4 slots per LDS buffer (>= 21*128)

// ---------------- workspace layout ----------------
static constexpr size_t alignUp16(size_t x) { return (x + 15) & ~(size_t)15; }
static constexpr size_t WS_MINE = 0;                                                // B*P f32
static constexpr size_t WS_CONF = WS_MINE + (size_t)BB * PP * sizeof(float);        // B*P u8
static constexpr size_t WS_BPI  = alignUp16(WS_CONF + (size_t)BB * PP);             // B*T i32
static constexpr size_t WS_LLP  = alignUp16(WS_BPI + (size_t)BB * TT * 4);          // B*24 f32
static constexpr size_t WS_NPP  = alignUp16(WS_LLP + (size_t)BB * NBLKX * 4);       // B*24 i32
static constexpr size_t WS_PCE  = alignUp16(WS_NPP + (size_t)BB * NBLKX * 4);       // B*24 f32
static constexpr size_t WS_LLB  = alignUp16(WS_PCE + (size_t)BB * NBLKX * 4);       // B f32
static constexpr size_t WS_LCB  = alignUp16(WS_LLB + (size_t)BB * 4);               // B f32
static constexpr size_t WS_NPB  = alignUp16(WS_LCB + (size_t)BB * 4);               // B i32
// total ~7.9 MB

// ---------------- CDNA5 async global->LDS helpers ----------------
typedef int v4i __attribute__((ext_vector_type(4)));
typedef __attribute__((address_space(1))) v4i* g_v4i_ptr;   // global (AS1) int4*
typedef __attribute__((address_space(3))) v4i* l_v4i_ptr;   // LDS (AS3) int4*
typedef __attribute__((address_space(3))) void as3_void;

__device__ __forceinline__ void async_b128(const void* g, void* l) {
#if defined(__HIP_DEVICE_COMPILE__)
#if __has_builtin(__builtin_amdgcn_global_load_async_to_lds_b128)
  // cpol = 1 -> TH_LOAD_NT (gfx12 CPol: TH[2:0]); 509MB read-once stream, keep it out of L2
  __builtin_amdgcn_global_load_async_to_lds_b128((g_v4i_ptr)g, (l_v4i_ptr)l, 0, 1);
#else
  unsigned loff = (unsigned)(unsigned long long)(as3_void*)l;
  asm volatile("global_load_async_to_lds_b128 %0, %1, off th:TH_LOAD_NT"
               :: "v"(loff), "v"((unsigned long long)g) : "memory");
#endif
#endif
}

#if defined(__HIP_DEVICE_COMPILE__) && __has_builtin(__builtin_amdgcn_s_wait_asynccnt)
#define WAIT_ASYNC(n) __builtin_amdgcn_s_wait_asynccnt(n)
#elif defined(__HIP_DEVICE_COMPILE__)
#define WAIT_ASYNC(n) asm volatile("s_wait_asynccnt " #n ::: "memory")
#else
#define WAIT_ASYNC(n)
#endif

// ---------------- small math helpers ----------------
__device__ __forceinline__ float iou_corner(float ax1, float ay1, float ax2, float ay2, float areaA,
                                            float bx1, float by1, float bx2, float by2, float areaB) {
  float ix = fmaxf(fminf(ax2, bx2) - fmaxf(ax1, bx1), 0.f);
  float iy = fmaxf(fminf(ay2, by2) - fmaxf(ay1, by1), 0.f);
  float inter = ix * iy;
  return inter / (areaA + areaB - inter);
}
__device__ __forceinline__ float smoothl1(float d) {
  float ad = fabsf(d);
  return (ad < 1.f) ? 0.5f * d * d : ad - 0.5f;
}

// ================= Kernel 3 (defined first so it leads the disasm): CE over C=81,
// async global->LDS double-buffered pipeline — the bandwidth-critical 509 MB stream =================
__global__ void __launch_bounds__(128) ceKernel(const float* __restrict__ conf_data,
                                                const unsigned char* __restrict__ conf_t,
                                                float* __restrict__ mine,
                                                float* __restrict__ pce_out) {
  __shared__ __align__(16) float sbuf[2][CE_CAP16 * 4];   // 2 x 43008 B double buffer
  __shared__ float rf[128];
  const int b = blockIdx.y, x = blockIdx.x, tid = threadIdx.x;
  const char* base = (const char*)(conf_data + (size_t)b * PP * CC);

  auto issue = [&](int t, int buf) {
    int rows = min(CE_ROWS, PP - t * CE_ROWS);            // always a multiple of 4
    int n16 = rows * CC / 4;                              // float4 count (16B aligned tile)
    const char* g = base + (size_t)t * CE_ROWS * CC * 4;
    for (int it = 0; it < CE_NLOAD; ++it) {               // exactly 21 async issues per wave
      int idx = it * 128 + tid;
      int gidx = (idx < n16) ? idx : (n16 - 1);           // clamp; garbage lands in pad rows
      async_b128(g + (size_t)gidx * 16, &sbuf[buf][idx * 4]);
    }
  };

  float pce = 0.f;
  issue(x, 0);
  for (int j = 0; j < CE_TILES_PER_BLK; ++j) {
    const int t = x + NBLKX * j;
    const int buf = j & 1;
    if (j < CE_TILES_PER_BLK - 1) {
      issue(x + NBLKX * (j + 1), buf ^ 1);                // prefetch next tile
      WAIT_ASYNC(21);                                     // previous tile complete (in-order)
    } else {
      WAIT_ASYNC(0);
    }
    __syncthreads();                                      // all waves' LDS writes visible

    int rows = min(CE_ROWS, PP - t * CE_ROWS);
    if (tid < rows) {
      const float* r = &sbuf[buf][tid * CC];              // stride 81 dwords: conflict-free
      int cls = conf_t[(size_t)b * PP + t * CE_ROWS + tid];
      float m = r[0];
      for (int c = 1; c < CC; ++c) m = fmaxf(m, r[c]);
      float s = 0.f;
      for (int c = 0; c < CC; ++c) s += __expf(r[c] - m);
      float ce = m + __logf(s) - r[cls];
      float mv;
      if (cls > 0) { pce += ce; mv = 0.f; } else mv = ce;
      mine[(size_t)b * PP + t * CE_ROWS + tid] = mv;
    }
    __syncthreads();                                      // buffer free before reuse at j+2
  }
  rf[tid] = pce;
  __syncthreads();
  for (int s = 64; s > 0; s >>= 1) {
    if (tid < s) rf[tid] += rf[tid + s];
    __syncthreads();
  }
  if (tid == 0) pce_out[b * NBLKX + x] = rf[0];
}

// ================= Kernel 1: best prior per (b,t) — argmax over P =================
__global__ void __launch_bounds__(256) bestPriorKernel(const float* __restrict__ priors,
                                                       const float* __restrict__ tboxes,
                                                       int* __restrict__ bpi_out) {
  const int t = blockIdx.x, b = blockIdx.y, tid = threadIdx.x;
  const float* tb = tboxes + ((size_t)b * TT + t) * 4;
  const float ax1 = tb[0], ay1 = tb[1], ax2 = tb[2], ay2 = tb[3];
  const float areaA = (ax2 - ax1) * (ay2 - ay1);
  float best = -1.f;
  int bidx = 0;
  const float4* pr4 = (const float4*)priors;
  for (int p = tid; p < PP; p += 256) {
    float4 c = pr4[p];
    float hw = 0.5f * c.z, hh = 0.5f * c.w;
    float ov = iou_corner(ax1, ay1, ax2, ay2, areaA,
                          c.x - hw, c.y - hh, c.x + hw, c.y + hh, c.z * c.w);
    if (ov > best) { best = ov; bidx = p; }   // ascending p -> first max kept
  }
  __shared__ float rv[256];
  __shared__ int ri[256];
  rv[tid] = best; ri[tid] = bidx;
  __syncthreads();
  for (int s = 128; s > 0; s >>= 1) {
    if (tid < s) {
      float v2 = rv[tid + s]; int i2 = ri[tid + s];
      if (v2 > rv[tid] || (v2 == rv[tid] && i2 < ri[tid])) { rv[tid] = v2; ri[tid] = i2; }
    }
    __syncthreads();
  }
  if (tid == 0) bpi_out[b * TT + t] = ri[0];
}

// ================= Kernel 2: per-prior match + encode + smooth-L1 =================
__global__ void __launch_bounds__(256) matchKernel(const float* __restrict__ priors,
                                                   const float* __restrict__ tboxes,
                                                   const int* __restrict__ tlabels,
                                                   const float* __restrict__ loc_data,
                                                   const int* __restrict__ bpi,
                                                   unsigned char* __restrict__ conf_t,
                                                   float* __restrict__ llp,
                                                   int* __restrict__ npp) {
  const int b = blockIdx.y, x = blockIdx.x, tid = threadIdx.x;
  __shared__ float sT[TT * 4];
  __shared__ int sL[TT];
  __shared__ int sB[TT];
  if (tid < TT * 4) sT[tid] = tboxes[(size_t)b * TT * 4 + tid];
  if (tid < TT) { sL[tid] = tlabels[b * TT + tid]; sB[tid] = bpi[b * TT + tid]; }
  __syncthreads();

  float lsum = 0.f;
  int np = 0;
  for (int i = 0; i < 4; ++i) {
    int p = x * 1024 + i * 256 + tid;
    if (p < PP) {
      float4 c = ((const float4*)priors)[p];
      float hw = 0.5f * c.z, hh = 0.5f * c.w;
      float bx1 = c.x - hw, by1 = c.y - hh, bx2 = c.x + hw, by2 = c.y + hh;
      float areaB = c.z * c.w;
      float mv = -1.f; int mi = 0;
      for (int t = 0; t < TT; ++t) {
        float ax1 = sT[4 * t], ay1 = sT[4 * t + 1], ax2 = sT[4 * t + 2], ay2 = sT[4 * t + 3];
        float areaA = (ax2 - ax1) * (ay2 - ay1);
        float ov = iou_corner(ax1, ay1, ax2, ay2, areaA, bx1, by1, bx2, by2, areaB);
        if (ov > mv) { mv = ov; mi = t; }     // first max -> smallest t (JAX argmax axis=0)
      }
      for (int t = 0; t < TT; ++t)            // forced-match override, last write wins
        if (sB[t] == p) { mi = t; mv = 2.f; }
      int conf = (mv < THRESH) ? 0 : (sL[mi] + 1);
      conf_t[(size_t)b * PP + p] = (unsigned char)conf;
      if (conf > 0) {
        ++np;
        float mx1 = sT[4 * mi], my1 = sT[4 * mi + 1], mx2 = sT[4 * mi + 2], my2 = sT[4 * mi + 3];
        float g0 = (0.5f * (mx1 + mx2) - c.x) / (VAR0 * c.z);
        float g1 = (0.5f * (my1 + my2) - c.y) / (VAR0 * c.w);
        float g2 = logf((mx2 - mx1) / c.z) / VAR1;
        float g3 = logf((my2 - my1) / c.w) / VAR1;
        float4 ld = ((const float4*)loc_data)[(size_t)b * PP + p];
        lsum += smoothl1(ld.x - g0) + smoothl1(ld.y - g1) +
                smoothl1(ld.z - g2) + smoothl1(ld.w - g3);
      }
    }
  }
  __shared__ float rf[256];
  __shared__ int rn[256];
  rf[tid] = lsum; rn[tid] = np;
  __syncthreads();
  for (int s = 128; s > 0; s >>= 1) {
    if (tid < s) { rf[tid] += rf[tid + s]; rn[tid] += rn[tid + s]; }
    __syncthreads();
  }
  if (tid == 0) { llp[b * NBLKX + x] = rf[0]; npp[b * NBLKX + x] = rn[0]; }
}

// ================= Kernel 4: per-batch radix-select (hard negative mining) =================
__global__ void __launch_bounds__(256) selectKernel(const float* __restrict__ mine,
                                                    const float* __restrict__ llp,
                                                    const int* __restrict__ npp,
                                                    const float* __restrict__ pce,
                                                    float* __restrict__ llb,
                                                    float* __restrict__ lcb,
                                                    int* __restrict__ npb) {
  const int b = blockIdx.x, tid = threadIdx.x;
  __shared__ unsigned hist[256];
  __shared__ float rf[256];
  __shared__ unsigned ru[256];
  __shared__ unsigned sPrefix;
  __shared__ int sKrem;
  __shared__ int sK;
  __shared__ float sPce;
  if (tid == 0) {
    float ll = 0.f, pc = 0.f; int np = 0;
    for (int j = 0; j < NBLKX; ++j) {
      ll += llp[b * NBLKX + j]; pc += pce[b * NBLKX + j]; np += npp[b * NBLKX + j];
    }
    llb[b] = ll; npb[b] = np;
    int k = 3 * np; if (k > PP - 1) k = PP - 1;
    sK = k; sKrem = k; sPrefix = 0u; sPce = pc;
  }
  __syncthreads();
  const int k = sK;
  float negsum = 0.f;
  if (k > 0) {
    const float* mb = mine + (size_t)b * PP;
    for (int pass = 0; pass < 4; ++pass) {
      const int shift = 24 - 8 * pass;
      const unsigned maskHi = (pass == 0) ? 0u : (0xFFFFFFFFu << (shift + 8));
      hist[tid] = 0u;
      __syncthreads();
      const unsigned pref = sPrefix;
      for (int p = tid; p < PP; p += 256) {
        unsigned u = __float_as_uint(mb[p]);               // all values >= 0 -> bits monotone
        if ((u & maskHi) == pref) atomicAdd(&hist[(u >> shift) & 255], 1u);
      }
      __syncthreads();
      if (tid == 0) {
        int krem = sKrem; unsigned cum = 0; int bin = 255;
        for (; bin > 0; --bin) {
          unsigned c = hist[bin];
          if (cum + c >= (unsigned)krem) break;
          cum += c;
        }
        sKrem = krem - (int)cum;
        sPrefix = sPrefix | ((unsigned)bin << shift);
      }
      __syncthreads();
    }
    const unsigned vb = sPrefix;                            // exact k-th largest (bits)
    const float v = __uint_as_float(vb);
    float sg = 0.f; unsigned cg = 0;
    for (int p = tid; p < PP; p += 256) {
      float xv = mb[p];
      if (__float_as_uint(xv) > vb) { sg += xv; ++cg; }
    }
    rf[tid] = sg; ru[tid] = cg;
    __syncthreads();
    for (int s = 128; s > 0; s >>= 1) {
      if (tid < s) { rf[tid] += rf[tid + s]; ru[tid] += ru[tid + s]; }
      __syncthreads();
    }
    negsum = rf[0] + (float)(k - (int)ru[0]) * v;           // ties handled exactly (sum-wise)
  }
  if (tid == 0) lcb[b] = sPce + negsum;
}

// ================= Kernel 5: final combine =================
__global__ void finalKernel(const float* __restrict__ llb, const float* __restrict__ lcb,
                            const int* __restrict__ npb, float* __restrict__ out) {
  if (threadIdx.x == 0 && blockIdx.x == 0) {
    float ll = 0.f, lc = 0.f; int np = 0;
    for (int b = 0; b < BB; ++b) { ll += llb[b]; lc += lcb[b]; np += npb[b]; }
    float n = (float)((np > 1) ? np : 1);
    out[0] = ll / n;
    out[1] = lc / n;
  }
}

// ================= host launcher =================
extern "C" void kernel_launch(void* const* d_in, const int* in_sizes, int n_in,
                              void* d_out, int out_size, void* d_ws, size_t ws_size,
                              hipStream_t stream) {
  (void)in_sizes; (void)n_in; (void)out_size; (void)ws_size;
  const float* loc    = (const float*)d_in[0];
  const float* conf   = (const float*)d_in[1];
  const float* priors = (const float*)d_in[2];
  const float* tbox   = (const float*)d_in[3];
  const int*   tlab   = (const int*)d_in[4];

  char* ws = (char*)d_ws;
  float* mine           = (float*)(ws + WS_MINE);
  unsigned char* conf_t = (unsigned char*)(ws + WS_CONF);
  int* bpi              = (int*)(ws + WS_BPI);
  float* llp            = (float*)(ws + WS_LLP);
  int* npp              = (int*)(ws + WS_NPP);
  float* pce            = (float*)(ws + WS_PCE);
  float* llb            = (float*)(ws + WS_LLB);
  float* lcb            = (float*)(ws + WS_LCB);
  int* npb              = (int*)(ws + WS_NPB);

  bestPriorKernel<<<dim3(TT, BB), 256, 0, stream>>>(priors, tbox, bpi);
  matchKernel<<<dim3(NBLKX, BB), 256, 0, stream>>>(priors, tbox, tlab, loc, bpi, conf_t, llp, npp);
  ceKernel<<<dim3(NBLKX, BB), 128, 0, stream>>>(conf, conf_t, mine, pce);
  selectKernel<<<BB, 256, 0, stream>>>(mine, llp, npp, pce, llb, lcb, npb);
  finalKernel<<<1, 32, 0, stream>>>(llb, lcb, npb, (float*)d_out);
}